// Linear_32315333935471
// MI455X (gfx1250) — compile-verified
//
#include <hip/hip_runtime.h>

// ---------------------------------------------------------------------------
// out = X @ R^T,  R = diag(b) * (I + 2Q + 2Q^2 + 2Q^3 + Q^4) * diag(a),
// Q = skew(w) clamped to Frobenius norm EPS.
//
// Phase 1 (tiny): build_q -> gemm128 (Q2=Q@Q) -> assemble_R (Q3=Q2@Q,
//                 Q4=Q2@Q2, fold series + b/a scaling, write R row-major).
// Phase 2 (HBM-bound, 64 MB traffic => ~2.75us floor @ 23.3 TB/s):
//                 X @ R^T via V_WMMA_F32_16X16X4_F32, one wave per 16-wide
//                 N column, all 32 B frags held in registers, X streamed
//                 non-temporally, out stored non-temporally (R stays in L2).
// ---------------------------------------------------------------------------

#define QSIZE 128
#define NW    (QSIZE * (QSIZE - 1) / 2)   // 8128
#define CN_EPS 0.9f

typedef __attribute__((ext_vector_type(2))) float v2f;
typedef __attribute__((ext_vector_type(8))) float v8f;

static __device__ __forceinline__ v8f wmma_f32x4(v2f a, v2f b, v8f c) {
  // D(16x16,f32) = A(16x4,f32) * B(4x16,f32) + C
  return __builtin_amdgcn_wmma_f32_16x16x4_f32(
      /*neg_a=*/false, a, /*neg_b=*/false, b,
      /*c_mod=*/(short)0, c, /*reuse_a=*/false, /*reuse_b=*/false);
}

// ---------------------------------------------------------------------------
// Kernel A: norm reduction + scaled skew-symmetric Q.  1 block x 256 threads.
// ---------------------------------------------------------------------------
__global__ __launch_bounds__(256) void build_q(const float* __restrict__ w,
                                               float* __restrict__ Q) {
  __shared__ float red[256];
  const int t = threadIdx.x;

  float s = 0.0f;
  for (int i = t; i < NW; i += 256) {
    float v = w[i];
    s += v * v;
  }
  red[t] = s;
  __syncthreads();
  for (int o = 128; o > 0; o >>= 1) {
    if (t < o) red[t] += red[t + o];
    __syncthreads();
  }
  const float norm = sqrtf(2.0f * red[0]);            // ||Q||_F of skew matrix
  const bool finite = (norm == norm) && (norm < 3.0e38f);
  const float scale = (finite && norm > CN_EPS) ? (CN_EPS / (norm + 1e-12f))
                                                : 1.0f;

  if (t < QSIZE) {
    const int r = t;
    const int base = r * (2 * QSIZE - r - 1) / 2;     // triu(k=1) row offset
    Q[r * QSIZE + r] = 0.0f;
    for (int c = r + 1; c < QSIZE; ++c) {
      const float v = w[base + (c - r - 1)] * scale;
      Q[r * QSIZE + c] = v;
      Q[c * QSIZE + r] = -v;
    }
  }
}

// ---------------------------------------------------------------------------
// Kernel B: C = A @ B, all 128x128 row-major.  Grid 8 x 256 threads
// (64 waves == 64 independent 16x16 tiles, K=128 -> 32 WMMA each).
// ---------------------------------------------------------------------------
__global__ __launch_bounds__(256) void gemm128(const float* __restrict__ A,
                                               const float* __restrict__ B,
                                               float* __restrict__ C) {
  const int lane = threadIdx.x & 31;
  const int wave = threadIdx.x >> 5;
  const int tile = blockIdx.x * 8 + wave;
  const int m0 = (tile >> 3) * 16;
  const int n0 = (tile & 7) * 16;
  const int lo = lane & 15, hi = lane >> 4;

  v8f acc = {};
#pragma unroll
  for (int kk = 0; kk < QSIZE; kk += 4) {
    // A frag: row m0+lo, cols kk+2*hi .. +1 (contiguous pair)
    v2f a = *(const v2f*)(A + (m0 + lo) * QSIZE + kk + 2 * hi);
    // B frag: rows kk+2*hi, kk+2*hi+1 at col n0+lo (stride-128 pair)
    const float* bp = B + (kk + 2 * hi) * QSIZE + n0 + lo;
    v2f b;
    b.x = bp[0];
    b.y = bp[QSIZE];
    acc = wmma_f32x4(a, b, acc);
  }
#pragma unroll
  for (int i = 0; i < 8; ++i)
    C[(m0 + i + 8 * hi) * QSIZE + n0 + lo] = acc[i];
}

// ---------------------------------------------------------------------------
// Kernel C: R = diag(vb) * (I + 2Q + 2Q2 + 2*(Q2@Q) + (Q2@Q2)) * diag(va).
// Same 64-wave tiling; two WMMA chains per tile share the Q2 A-fragments.
// ---------------------------------------------------------------------------
__global__ __launch_bounds__(256) void assemble_R(const float* __restrict__ Q,
                                                  const float* __restrict__ Q2,
                                                  const float* __restrict__ vb,
                                                  const float* __restrict__ va,
                                                  float* __restrict__ R) {
  const int lane = threadIdx.x & 31;
  const int wave = threadIdx.x >> 5;
  const int tile = blockIdx.x * 8 + wave;
  const int m0 = (tile >> 3) * 16;
  const int n0 = (tile & 7) * 16;
  const int lo = lane & 15, hi = lane >> 4;

  v8f acc3 = {};  // Q2 @ Q
  v8f acc4 = {};  // Q2 @ Q2
#pragma unroll
  for (int kk = 0; kk < QSIZE; kk += 4) {
    v2f a = *(const v2f*)(Q2 + (m0 + lo) * QSIZE + kk + 2 * hi);
    const float* bq = Q + (kk + 2 * hi) * QSIZE + n0 + lo;
    const float* b2 = Q2 + (kk + 2 * hi) * QSIZE + n0 + lo;
    v2f b3, b4;
    b3.x = bq[0];
    b3.y = bq[QSIZE];
    b4.x = b2[0];
    b4.y = b2[QSIZE];
    acc3 = wmma_f32x4(a, b3, acc3);
    acc4 = wmma_f32x4(a, b4, acc4);
  }

  const float an = va[n0 + lo];
#pragma unroll
  for (int i = 0; i < 8; ++i) {
    const int row = m0 + i + 8 * hi;
    const int col = n0 + lo;
    const float qv = Q[row * QSIZE + col];
    const float q2v = Q2[row * QSIZE + col];
    float r = ((row == col) ? 1.0f : 0.0f) + 2.0f * qv + 2.0f * q2v +
              2.0f * acc3[i] + acc4[i];
    R[row * QSIZE + col] = vb[row] * r * an;
  }
}

// ---------------------------------------------------------------------------
// Main kernel: out = X @ R^T.  Block = 256 rows x 128 cols; 8 waves, each
// owns a 16-wide N column.  B frags (rows of R, since B = R^T) preloaded into
// 64 VGPRs as contiguous float2; then 16 M-tiles of 32 WMMA streaming X.
// X loads / out stores are non-temporal (single-use streams); R stays hot.
// ---------------------------------------------------------------------------
__global__ __launch_bounds__(256) void xr_main(const float* __restrict__ X,
                                               const float* __restrict__ R,
                                               float* __restrict__ out) {
  const int lane = threadIdx.x & 31;
  const int wave = threadIdx.x >> 5;
  const int lo = lane & 15, hi = lane >> 4;
  const int n0 = wave * 16;
  const int row_base = blockIdx.x * 256;

  // Preload B fragments: B[k][n] = R[n][k] -> contiguous pairs along k.
  // Regular-temporal loads: R is reused by every block, keep it cached.
  v2f bf[32];
#pragma unroll
  for (int s = 0; s < 32; ++s)
    bf[s] = *(const v2f*)(R + (n0 + lo) * QSIZE + 4 * s + 2 * hi);

  for (int mt = 0; mt < 16; ++mt) {
    const int m0 = row_base + mt * 16;
    const float* xrow = X + (m0 + lo) * QSIZE + 2 * hi;
    v8f acc = {};
#pragma unroll
    for (int s = 0; s < 32; ++s) {
      // Non-temporal: X is streamed exactly once.
      v2f a = __builtin_nontemporal_load((const v2f*)(xrow + 4 * s));
      acc = wmma_f32x4(a, bf[s], acc);
    }
    float* orow = out + (m0 + 8 * hi) * QSIZE + n0 + lo;
#pragma unroll
    for (int i = 0; i < 8; ++i)
      __builtin_nontemporal_store(acc[i], orow + i * QSIZE);
  }
}

// ---------------------------------------------------------------------------
// Launcher.  Inputs: x (8*8192*128 f32), w (8128 f32), vector_b (128),
// vector_a (128).  Workspace: Q | Q2 | R  (3 * 64 KB).
// ---------------------------------------------------------------------------
extern "C" void kernel_launch(void* const* d_in, const int* in_sizes, int n_in,
                              void* d_out, int out_size, void* d_ws,
                              size_t ws_size, hipStream_t stream) {
  const float* x = (const float*)d_in[0];
  const float* w = (const float*)d_in[1];
  const float* vb = (const float*)d_in[2];
  const float* va = (const float*)d_in[3];
  float* out = (float*)d_out;

  float* Q = (float*)d_ws;
  float* Q2 = Q + QSIZE * QSIZE;
  float* R = Q2 + QSIZE * QSIZE;

  const int rows = in_sizes[0] / QSIZE;       // 65536
  const int mblocks = rows / 256;             // 256

  build_q<<<1, 256, 0, stream>>>(w, Q);
  gemm128<<<8, 256, 0, stream>>>(Q, Q, Q2);
  assemble_R<<<8, 256, 0, stream>>>(Q, Q2, vb, va, R);
  xr_main<<<mblocks, 256, 0, stream>>>(x, R, out);
}